// LengthRegulator_20143396618582
// MI455X (gfx1250) — compile-verified
//
#include <hip/hip_runtime.h>
#include <stdint.h>

// LengthRegulator: B=64, C=384, T=512, MAX_NEW_TIME=4096, DUR_MAX=8.
// Bandwidth-bound: ~402MB stores + ~50MB reads @ 23.3 TB/s => ~20us floor.
//   kernel A (gather, listed first for disasm visibility): per (batch, j-tile)
//     block; Tensor Data Mover double-buffers 16-row x tiles (32KB each) into
//     LDS (tensor_load_to_lds + s_wait_tensorcnt pipelining), gather with
//     register-held idx quads, coalesced float4 stores.
//   kernel B (scan): per-batch LDS scan of durations + scatter-inverse of
//     searchsorted (idx[j] in [0,T) or -1 when j >= total), plus new_mask.

#define B_ 64
#define C_ 384
#define T_ 512
#define M_ 4096
#define CH 16                  // rows per TDM tile: 16*512*4 = 32KB
#define NCHUNK (C_ / CH)       // 24

typedef uint32_t u32x4 __attribute__((ext_vector_type(4)));
typedef int32_t  i32x4 __attribute__((ext_vector_type(4)));
typedef int32_t  i32x8 __attribute__((ext_vector_type(8)));

// Only the DEVICE pass sees AMDGCN builtins; the host pass parses the fallback.
#if defined(__HIP_DEVICE_COMPILE__)
#if !__has_builtin(__builtin_amdgcn_tensor_load_to_lds) || \
    !__has_builtin(__builtin_amdgcn_s_wait_tensorcnt)
#error "device pass: CDNA5 TDM builtins unavailable - switch to inline asm tensor_load_to_lds"
#endif
#define HAVE_TDM 1
#endif

#if defined(HAVE_TDM)
// Issue one TDM load of a 2D tile: CH rows x T_ floats, row stride T_ floats,
// from global gptr into LDS byte offset lds_off. Descriptor per CDNA5 ISA:
//  group0: count=1 | lds_addr | global_addr(57b) | type=2
//  group1: data_size=2(4B) | tensor_dim0=T_ | tensor_dim1=CH | tile_dim0=T_
//          | tile_dim1=CH | tensor_dim0_stride=T_
__device__ __forceinline__ void tdm_load_rows(const float* gptr, uint32_t lds_off) {
  uint64_t ga = (uint64_t)(uintptr_t)gptr;
  u32x4 g0;
  g0.x = 1u;                                                // count=1 (valid D#)
  g0.y = lds_off;                                           // lds_addr (bytes)
  g0.z = (uint32_t)ga;                                      // global_addr[31:0]
  g0.w = (uint32_t)((ga >> 32) & 0x01FFFFFFu) | (2u << 30); // addr[56:32] | type=2
  i32x8 g1;
  g1[0] = (int)(2u << 16);             // data_size=2 -> 4 bytes
  g1[1] = (int)((T_ & 0xFFFF) << 16);  // tensor_dim0[15:0]=512 (atomic_barrier=0)
  g1[2] = (int)((CH & 0xFFFF) << 16);  // tensor_dim0[31:16]=0 | tensor_dim1=CH
  g1[3] = (int)((T_ & 0xFFFF) << 16);  // tensor_dim1 hi=0 | tile_dim0=512
  g1[4] = CH;                          // tile_dim1=CH, tile_dim2=0
  g1[5] = T_;                          // tensor_dim0_stride[31:0]=512
  g1[6] = 0;                           // stride hi | tensor_dim1_stride lo
  g1[7] = 0;
  i32x4 gz = {0, 0, 0, 0};
#if __clang_major__ >= 23
  i32x8 gz8 = {0, 0, 0, 0, 0, 0, 0, 0};
  __builtin_amdgcn_tensor_load_to_lds(g0, g1, gz, gz, gz8, 0);
#else
  __builtin_amdgcn_tensor_load_to_lds(g0, g1, gz, gz, 0);
#endif
}
#endif // HAVE_TDM

// ---------------------------------------------------------------------------
// Gather kernel: grid (M_/512, B). Block = 256 threads: thread owns a j-quad
// (q = tid&127 -> j = j0+4q..j0+4q+3) and half of the CH rows (rh = tid>>7).
// TDM double-buffers the 24 channel chunks; wave 0 drives the DMA pipeline.
__global__ void __launch_bounds__(256) lr_gather_kernel(
    const float* __restrict__ x, const int* __restrict__ idx_ws,
    float* __restrict__ out) {
  __shared__ float xs[2][CH * T_];   // 64KB double buffer
  const int b   = blockIdx.y;
  const int j0  = blockIdx.x * 512;
  const int tid = threadIdx.x;
  const int q   = tid & 127;
  const int rh  = tid >> 7;

  // Per-thread gather indices (monotonic non-decreasing across lanes)
  const int* ip = idx_ws + (size_t)b * M_ + j0 + 4 * q;
  const int id0 = ip[0], id1 = ip[1], id2 = ip[2], id3 = ip[3];

  const float* xb = x + (size_t)b * C_ * T_;
  float* ob = out + (size_t)b * C_ * M_ + j0 + 4 * q;

#if defined(HAVE_TDM)
  const uint32_t lds_lo = (uint32_t)(uintptr_t)(&xs[0][0]);
  const uint32_t lds_hi = (uint32_t)(uintptr_t)(&xs[1][0]);
  if (tid < 32) {                       // prime the pipeline (wave 0 only)
    tdm_load_rows(xb, lds_lo);
  }
#endif

  for (int i = 0; i < NCHUNK; ++i) {
#if defined(HAVE_TDM)
    if (tid < 32) {
      if (i + 1 < NCHUNK) {
        // issue next chunk's DMA, then retire the current one (in-order TDM):
        tdm_load_rows(xb + (size_t)(i + 1) * CH * T_,
                      ((i + 1) & 1) ? lds_hi : lds_lo);
        __builtin_amdgcn_s_wait_tensorcnt(1);
      } else {
        __builtin_amdgcn_s_wait_tensorcnt(0);
      }
    }
#else
    {  // host-pass / no-TDM fallback: plain coalesced staging
      const float* xg = xb + (size_t)i * CH * T_;
      float* dstb = &xs[i & 1][0];
      for (int k = tid; k < CH * T_; k += 256) dstb[k] = xg[k];
    }
#endif
    __syncthreads();                    // buffer i is ready for all waves
    const float* buf = xs[i & 1];
    const int c0 = i * CH;
    for (int r = rh; r < CH; r += 2) {
      const float* row = buf + r * T_;
      float4 v;
      v.x = (id0 >= 0) ? row[id0] : 0.0f;
      v.y = (id1 >= 0) ? row[id1] : 0.0f;
      v.z = (id2 >= 0) ? row[id2] : 0.0f;
      v.w = (id3 >= 0) ? row[id3] : 0.0f;
      *reinterpret_cast<float4*>(ob + (size_t)(c0 + r) * M_) = v;
    }
    __syncthreads();                    // done reading buf before DMA reuses it
  }
}

// ---------------------------------------------------------------------------
// Scan kernel: one block per batch. Inclusive scan of durations; invert the
// searchsorted by scattering (position p in [csum[t-1], csum[t]) -> idx=t);
// positions >= total stay -1. Emits idx table to workspace and new_mask.
__global__ void __launch_bounds__(T_) lr_scan_kernel(
    const int* __restrict__ dur, const float* __restrict__ x,
    int* __restrict__ idx_ws, float* __restrict__ mask_out) {
  __shared__ int s[T_];
  __shared__ int idxs[M_];
  const int b = blockIdx.x;
  const int t = threadIdx.x;

  const int d = dur[(size_t)b * T_ + t];
  s[t] = d;
  __syncthreads();
  // Hillis-Steele inclusive scan (9 steps)
  for (int off = 1; off < T_; off <<= 1) {
    int v = (t >= off) ? s[t - off] : 0;
    __syncthreads();
    s[t] += v;
    __syncthreads();
  }
  for (int i = t; i < M_; i += T_) idxs[i] = -1;
  __syncthreads();
  const int start = s[t] - d;  // exclusive prefix
  for (int k = 0; k < d; ++k) {
    int p = start + k;
    if (p < M_) idxs[p] = t;
  }
  __syncthreads();

  const float* xrow0 = x + (size_t)b * C_ * T_;  // channel 0 row
  for (int i = t; i < M_; i += T_) {
    int id = idxs[i];
    idx_ws[(size_t)b * M_ + i] = id;
    float xv = (id >= 0) ? xrow0[id] : 0.0f;
    // new_mask = (out[:,0,:] != 0); emitted as 0.0/1.0 in the f32 output buffer
    mask_out[(size_t)b * M_ + i] = (xv != 0.0f) ? 1.0f : 0.0f;
  }
}

// ---------------------------------------------------------------------------
extern "C" void kernel_launch(void* const* d_in, const int* in_sizes, int n_in,
                              void* d_out, int out_size, void* d_ws, size_t ws_size,
                              hipStream_t stream) {
  (void)in_sizes; (void)n_in; (void)out_size; (void)ws_size;
  const float* x   = (const float*)d_in[0];   // (B, C, T) f32
  const int*   dur = (const int*)d_in[1];     // (B, 1, T) i32
  float* out  = (float*)d_out;                        // (B, C, M) f32
  float* mask = out + (size_t)B_ * C_ * M_;           // (B, 1, M) as 0.0/1.0
  int*   idx_ws = (int*)d_ws;                         // (B, M) i32, 1 MB

  lr_scan_kernel<<<dim3(B_), dim3(T_), 0, stream>>>(dur, x, idx_ws, mask);
  lr_gather_kernel<<<dim3(M_ / 512, B_), dim3(256), 0, stream>>>(x, idx_ws, out);
}